// Model_51453708206389
// MI455X (gfx1250) — compile-verified
//
#include <hip/hip_runtime.h>
#include <math.h>

typedef __attribute__((ext_vector_type(2))) float v2f;
typedef __attribute__((ext_vector_type(4))) float v4f;
typedef __attribute__((ext_vector_type(8))) float v8f;

// Matches the builtin's parameter type: int __attribute__((vector_size(16)))
typedef int vi4 __attribute__((vector_size(16)));
typedef __attribute__((address_space(1))) vi4 as1_vi4;  // global
typedef __attribute__((address_space(3))) vi4 as3_vi4;  // LDS

#define B_   8
#define N1_  32
#define N2_  8
#define G_   4
#define D_   128
#define S2_  16384
#define SP_  2048
#define PARTS 8
#define WAVES 4
#define THREADS (WAVES * 32)
#define KEYS_PER_PART (SP_ / PARTS)            // 256
#define KEYS_PER_WAVE (KEYS_PER_PART / WAVES)  // 64
#define CHUNKS (KEYS_PER_WAVE / 16)            // 4
#define PART_STRIDE (8 + G_ * D_)              // m[4], l[4], O[4][128] = 520 floats

#if __has_builtin(__builtin_amdgcn_global_load_async_to_lds_b128)
#define USE_ASYNC_LDS 1
#else
#define USE_ASYNC_LDS 0
#endif

#if USE_ASYNC_LDS
#define ASYNC_CP_B128(gsrc, ldst)                                        \
  __builtin_amdgcn_global_load_async_to_lds_b128((as1_vi4*)(gsrc),       \
                                                 (as3_vi4*)(ldst), 0, 0)
#if __has_builtin(__builtin_amdgcn_s_wait_asynccnt)
#define WAIT_ASYNC(n) __builtin_amdgcn_s_wait_asynccnt(n)
#else
#define WAIT_ASYNC(n) asm volatile("s_wait_asynccnt %0" ::"i"(n) : "memory")
#endif
#else
#define WAIT_ASYNC(n)
#endif

// Gather 16 K rows + 16 V rows (512 B each) for one chunk into per-wave LDS
// tiles. Async path: 32 GLOBAL_LOAD_ASYNC_TO_LDS_B128 in flight (ASYNCcnt).
__device__ __forceinline__ void stage_chunk(const float* __restrict__ kk,
                                            const float* __restrict__ vv,
                                            size_t base_b, int n2, int myidx,
                                            int lane, float* Ktb, float* Vtb) {
#if USE_ASYNC_LDS
#pragma unroll
  for (int r = 0; r < 16; ++r) {
    const int row = __shfl(myidx, r, 32);
    const size_t rb = ((base_b + row) * N2_ + n2) * D_ + lane * 4;
    ASYNC_CP_B128(kk + rb, Ktb + r * D_ + lane * 4);
    ASYNC_CP_B128(vv + rb, Vtb + r * D_ + lane * 4);
  }
#else
  // Fallback: batch 16 b128 loads before any ds_store for memory parallelism.
  v4f kreg[8], vreg[8];
#pragma unroll
  for (int half = 0; half < 2; ++half) {
#pragma unroll
    for (int r = 0; r < 8; ++r) {
      const int row = __shfl(myidx, half * 8 + r, 32);
      const size_t rb = ((base_b + row) * N2_ + n2) * D_;
      kreg[r] = ((const v4f*)(kk + rb))[lane];
      vreg[r] = ((const v4f*)(vv + rb))[lane];
    }
#pragma unroll
    for (int r = 0; r < 8; ++r) {
      *(v4f*)&Ktb[(half * 8 + r) * D_ + lane * 4] = kreg[r];
      *(v4f*)&Vtb[(half * 8 + r) * D_ + lane * 4] = vreg[r];
    }
  }
#endif
}

// ---------------------------------------------------------------------------
// Stage 1: one block per (b, n2, partition). Each wave handles 64 keys in
// 16-key chunks (double-buffered async gather) with online softmax; QK and
// PV via v_wmma_f32_16x16x4_f32.
// ---------------------------------------------------------------------------
__global__ void sparse_attn_stage1(const float* __restrict__ q,
                                   const float* __restrict__ kk,
                                   const float* __restrict__ vv,
                                   const int*   __restrict__ sidx,
                                   const float* __restrict__ scale_p,
                                   float*       __restrict__ part_out) {
  const int unit = blockIdx.x / PARTS;   // 0..63  -> (b, n2)
  const int part = blockIdx.x % PARTS;
  const int b    = unit / N2_;
  const int n2   = unit % N2_;
  const int tid  = threadIdx.x;
  const int lane = tid & 31;
  const int wave = tid >> 5;
  const int col  = lane & 15;   // N index (head for QK-D / PV-A; dim for PV-B)
  const int hi   = lane >> 4;   // lane-group: splits the K (reduction) dim

  extern __shared__ char smem_raw[];
  float* Qs = (float*)smem_raw;                 // [4][128]
  float* Kt = Qs + G_ * D_;                     // [WAVES][2][16][128]
  float* Vt = Kt + WAVES * 2 * 16 * D_;         // [WAVES][2][16][128]
  float* Pt = Vt + WAVES * 2 * 16 * D_;         // [WAVES][16][16]  Pt[h][j]
  float* Om = Pt + WAVES * 256;                 // [WAVES][4][128]
  float* Ml = Om + WAVES * G_ * D_;             // [WAVES][4][2]
  float* Ktw = Kt + wave * 2 * 16 * D_;
  float* Vtw = Vt + wave * 2 * 16 * D_;
  float* Ptw = Pt + wave * 256;

  // Load the 4-head query tile for this kv-head group.
  for (int i = tid; i < G_ * D_; i += THREADS) {
    const int h = i >> 7, d = i & 127;
    Qs[i] = q[((size_t)b * N1_ + (n2 * G_ + h)) * D_ + d];
  }
  __syncthreads();
  const float scale = scale_p[0];

  float m_run = -__builtin_inff();
  float l_run = 0.0f;
  const v8f vzero = {0.f, 0.f, 0.f, 0.f, 0.f, 0.f, 0.f, 0.f};
  v8f accO[8];
#pragma unroll
  for (int n = 0; n < 8; ++n) accO[n] = vzero;

  const int keybase = part * KEYS_PER_PART + wave * KEYS_PER_WAVE;
  const int sbase   = (b * N2_ + n2) * SP_;
  const size_t base_b = (size_t)b * S2_;

  // Prologue: stage chunk 0 into buffer 0.
  {
    int idx0 = 0;
    if (lane < 16) idx0 = sidx[sbase + keybase + lane];
    stage_chunk(kk, vv, base_b, n2, idx0, lane, Ktw, Vtw);
  }

  for (int c = 0; c < CHUNKS; ++c) {
    const int cur = c & 1;
    if (c + 1 < CHUNKS) {
      // Issue next chunk's gather into the other buffer, then wait until only
      // those 32 async ops remain outstanding (in-order => chunk c resident).
      int idxn = 0;
      if (lane < 16) idxn = sidx[sbase + keybase + (c + 1) * 16 + lane];
      stage_chunk(kk, vv, base_b, n2, idxn, lane,
                  Ktw + (1 - cur) * 16 * D_, Vtw + (1 - cur) * 16 * D_);
      WAIT_ASYNC(32);
    } else {
      WAIT_ASYNC(0);
    }
    float* Ktc = Ktw + cur * 16 * D_;
    float* Vtc = Vtw + cur * 16 * D_;

    // ---- QK^T: D[key][head] += sum_d K[key][d] * Q[head][d] --------------
    // A (16x4): lane=M(key), vgprs = K[ M ][ 4s + 2*hi + {0,1} ]
    // B (4x16): lane=N(head), vgprs = Q[ N ][ 4s + 2*hi + {0,1} ]
    v8f s = vzero;
#pragma unroll
    for (int st = 0; st < 32; ++st) {
      const int d0 = st * 4 + 2 * hi;
      const v2f a  = *(const v2f*)&Ktc[col * D_ + d0];
      const v2f bq = *(const v2f*)&Qs[(col & 3) * D_ + d0];  // cols>=4 dup head col&3 (rows ignored later)
      s = __builtin_amdgcn_wmma_f32_16x16x4_f32(false, a, false, bq,
                                                (short)0, s, false, false);
    }

    // ---- Online softmax over the 16-key chunk (per head = column) --------
    float cm = -__builtin_inff();
#pragma unroll
    for (int r = 0; r < 8; ++r) { s[r] *= scale; cm = fmaxf(cm, s[r]); }
    cm = fmaxf(cm, __shfl_xor(cm, 16, 32));          // combine key groups 0-7 / 8-15
    const float mnew = fmaxf(m_run, cm);
    const float f    = __expf(m_run - mnew);
    m_run = mnew;
    float fr[4];
#pragma unroll
    for (int h = 0; h < 4; ++h) fr[h] = __shfl(f, h, 32);  // per-head rescale factors

    float p[8];
    float psum = 0.f;
#pragma unroll
    for (int r = 0; r < 8; ++r) { p[r] = __expf(s[r] - m_run); psum += p[r]; }
    l_run = l_run * f + (psum + __shfl_xor(psum, 16, 32));

#pragma unroll
    for (int n = 0; n < 8; ++n)
#pragma unroll
      for (int r = 0; r < 8; ++r) accO[n][r] *= fr[r & 3];  // rows 0..3 are real heads

    // ---- Stash p tile to LDS in [head][key] layout -----------------------
    {
      const v4f p0 = {p[0], p[1], p[2], p[3]};
      const v4f p1 = {p[4], p[5], p[6], p[7]};
      *(v4f*)&Ptw[col * 16 + hi * 8 + 0] = p0;   // keys hi*8 .. hi*8+7
      *(v4f*)&Ptw[col * 16 + hi * 8 + 4] = p1;
    }

    // ---- P·V: O[head][dim] += sum_j p[head][j] * V[j][dim] ---------------
    // A (16x4): lane=M(head), vgprs = p[M][ 4kc + 2*hi + {0,1} ]
    // B (4x16): lane=N(dim),  vgprs = V[ 4kc + 2*hi + {0,1} ][ n*16 + N ]
#pragma unroll
    for (int kc = 0; kc < 4; ++kc) {
      const v2f ap = *(const v2f*)&Ptw[col * 16 + kc * 4 + 2 * hi];
      const int kr = kc * 4 + 2 * hi;
#pragma unroll
      for (int n = 0; n < 8; ++n) {
        v2f bv;
        bv[0] = Vtc[(kr + 0) * D_ + n * 16 + col];
        bv[1] = Vtc[(kr + 1) * D_ + n * 16 + col];
        accO[n] = __builtin_amdgcn_wmma_f32_16x16x4_f32(false, ap, false, bv,
                                                        (short)0, accO[n],
                                                        false, false);
      }
    }
  }

  // ---- Per-wave partials to LDS ------------------------------------------
  if (hi == 0) {   // D-tile rows 0..3 (heads) live in vgprs 0..3 of lanes 0..15
#pragma unroll
    for (int n = 0; n < 8; ++n)
#pragma unroll
      for (int r = 0; r < 4; ++r)
        Om[(wave * 4 + r) * D_ + n * 16 + col] = accO[n][r];
  }
  if (lane < 4) {  // column h = lane holds head-h running stats
    Ml[wave * 8 + lane * 2 + 0] = m_run;
    Ml[wave * 8 + lane * 2 + 1] = l_run;
  }
  __syncthreads();

  // ---- Combine waves of this block, emit partition partial ---------------
  float* dst = part_out + (size_t)(unit * PARTS + part) * PART_STRIDE;
  for (int i = tid; i < G_ * D_; i += THREADS) {
    const int h = i >> 7, d = i & 127;
    float mb = -__builtin_inff();
    for (int w = 0; w < WAVES; ++w) mb = fmaxf(mb, Ml[w * 8 + h * 2]);
    float lb = 0.f, ob = 0.f;
    for (int w = 0; w < WAVES; ++w) {
      const float e = __expf(Ml[w * 8 + h * 2] - mb);
      lb += Ml[w * 8 + h * 2 + 1] * e;
      ob += Om[(w * 4 + h) * D_ + d] * e;
    }
    dst[8 + i] = ob;
    if (d == 0) { dst[h] = mb; dst[4 + h] = lb; }
  }
}

// ---------------------------------------------------------------------------
// Stage 2: combine the PARTS flash partials per (b, n2) and normalize.
// ---------------------------------------------------------------------------
__global__ void sparse_attn_stage2(const float* __restrict__ part_in,
                                   float* __restrict__ out) {
  const int unit = blockIdx.x;           // 0..63
  const int b = unit / N2_, n2 = unit % N2_;
  const float* base = part_in + (size_t)unit * PARTS * PART_STRIDE;
  for (int i = threadIdx.x; i < G_ * D_; i += blockDim.x) {
    const int h = i >> 7, d = i & 127;
    float m = -__builtin_inff();
    for (int p = 0; p < PARTS; ++p)
      m = fmaxf(m, base[p * PART_STRIDE + h]);
    float l = 0.f, o = 0.f;
    for (int p = 0; p < PARTS; ++p) {
      const float e = __expf(base[p * PART_STRIDE + h] - m);
      l += base[p * PART_STRIDE + 4 + h] * e;
      o += base[p * PART_STRIDE + 8 + i] * e;
    }
    out[((size_t)b * N1_ + (n2 * G_ + h)) * D_ + d] = o / l;
  }
}

extern "C" void kernel_launch(void* const* d_in, const int* in_sizes, int n_in,
                              void* d_out, int out_size, void* d_ws, size_t ws_size,
                              hipStream_t stream) {
  const float* q     = (const float*)d_in[0];
  const float* k     = (const float*)d_in[1];
  const float* v     = (const float*)d_in[2];
  const int*   sidx  = (const int*)d_in[3];
  const float* scale = (const float*)d_in[4];
  float* ws = (float*)d_ws;

  const size_t smem_bytes =
      (size_t)(G_ * D_                     // Qs
               + 2 * WAVES * 2 * 16 * D_   // Kt + Vt (double buffered)
               + WAVES * 256               // Pt
               + WAVES * G_ * D_           // Om
               + WAVES * 8)                // Ml
      * sizeof(float);                     // ~142 KB < 320 KB WGP LDS

  (void)hipFuncSetAttribute((const void*)sparse_attn_stage1,
                            hipFuncAttributeMaxDynamicSharedMemorySize,
                            (int)smem_bytes);

  sparse_attn_stage1<<<B_ * N2_ * PARTS, THREADS, smem_bytes, stream>>>(
      q, k, v, sidx, scale, ws);
  sparse_attn_stage2<<<B_ * N2_, 128, 0, stream>>>(ws, (float*)d_out);
}